// Group2_23931557773507
// MI455X (gfx1250) — compile-verified
//
#include <hip/hip_runtime.h>
#include <hip/hip_bf16.h>

// Problem constants (match reference)
#define BATCH 32
#define NPTS  8192
#define NGRP  512
#define GSIZE 32
#define CTILES 32   // column tiles (of 16 points) swept per wave in dist_kernel

typedef __attribute__((ext_vector_type(2))) float v2f;
typedef __attribute__((ext_vector_type(8))) float v8f;

// ---------------------------------------------------------------------------
// Kernel 1: farthest point sampling. One block per batch, 1024 threads.
// Points live in registers (8 per thread, strided); 511 sequential steps of
// min-update + block argmax (tie -> smallest index, matching jnp.argmax).
// Writes center [B,G,3] and fps_idx (as float) [B,G] into d_out.
// ---------------------------------------------------------------------------
__global__ __launch_bounds__(1024)
void fps_kernel(const float* __restrict__ xyz, float* __restrict__ out) {
    const int b = blockIdx.x;
    const int t = threadIdx.x;
    const float* base = xyz + (size_t)b * NPTS * 3;

    float X[8], Y[8], Z[8], D[8];
#pragma unroll
    for (int i = 0; i < 8; ++i) {
        const int n = t + i * 1024;
        X[i] = base[n * 3 + 0];
        Y[i] = base[n * 3 + 1];
        Z[i] = base[n * 3 + 2];
        D[i] = __builtin_inff();
    }

    __shared__ float sd[1024];
    __shared__ int   si[1024];
    __shared__ int   slast;

    float* centerOut = out + (size_t)BATCH * NGRP * GSIZE * 3 + (size_t)b * NGRP * 3;
    float* idxOut    = out + (size_t)BATCH * NGRP * GSIZE * 3 + (size_t)BATCH * NGRP * 3
                           + (size_t)b * NGRP;

    if (t == 0) {
        centerOut[0] = base[0];
        centerOut[1] = base[1];
        centerOut[2] = base[2];
        idxOut[0]    = 0.0f;
    }

    int last = 0;
    for (int s = 1; s < NGRP; ++s) {
        const float lx = base[last * 3 + 0];
        const float ly = base[last * 3 + 1];
        const float lz = base[last * 3 + 2];

        float bd = -1.0f;
        int   bi = 0x7fffffff;
#pragma unroll
        for (int i = 0; i < 8; ++i) {
            const float dx = X[i] - lx;
            const float dy = Y[i] - ly;
            const float dz = Z[i] - lz;
            float d = dx * dx + dy * dy + dz * dz;
            d = fminf(D[i], d);
            D[i] = d;
            const int n = t + i * 1024;
            if (d > bd || (d == bd && n < bi)) { bd = d; bi = n; }
        }
        sd[t] = bd;
        si[t] = bi;
        __syncthreads();
        for (int off = 512; off > 0; off >>= 1) {
            if (t < off) {
                const float od = sd[t + off];
                const int   oi = si[t + off];
                if (od > sd[t] || (od == sd[t] && oi < si[t])) { sd[t] = od; si[t] = oi; }
            }
            __syncthreads();
        }
        if (t == 0) {
            const int w = si[0];
            slast = w;
            idxOut[s] = (float)w;
            const float* p = base + (size_t)w * 3;
            centerOut[s * 3 + 0] = p[0];
            centerOut[s * 3 + 1] = p[1];
            centerOut[s * 3 + 2] = p[2];
        }
        __syncthreads();
        last = slast;
    }
}

// ---------------------------------------------------------------------------
// Kernel 2: precompute |c|^2 per center. One block per batch.
// ---------------------------------------------------------------------------
__global__ void prep_kernel(const float* __restrict__ centers,
                            float* __restrict__ c2) {
    const int b = blockIdx.x;
    const int t = threadIdx.x;
    const float* cb = centers + (size_t)b * NGRP * 3;
    for (int g = t; g < NGRP; g += 256) {
        const float x = cb[g * 3 + 0], y = cb[g * 3 + 1], z = cb[g * 3 + 2];
        c2[(size_t)b * NGRP + g] = x * x + y * y + z * z;
    }
}

// ---------------------------------------------------------------------------
// Kernel 3: dist^2 slab via V_WMMA_F32_16X16X4_F32.
// dist2[g][n] = c2[g] + x2[n] - 2 * (C . X). Each wave owns one 16-row tile
// and sweeps CTILES consecutive 16-point column tiles: A (centers) and c2
// are loaded once and reused across the whole sweep. Per tile: one b96 point
// load per lane (no divergent control flow -- operands built with selects),
// x2 computed inline from the loaded coords, one WMMA, fused epilogue.
// A (16x4): lane = center row, half-wave picks K-pair {x,y} / {z,0}.
// B (4x16): lane = point column, same K striping.
// ---------------------------------------------------------------------------
__global__ void dist_kernel(const float* __restrict__ xyz,
                            const float* __restrict__ centers,
                            const float* __restrict__ c2,
                            float* __restrict__ slab,
                            int b, int g0) {
    const int lane = threadIdx.x & 31;
    const int wv   = threadIdx.x >> 5;
    const int wid  = blockIdx.x * 8 + wv;

    const int nchunks = NPTS / (16 * CTILES);     // 16 column chunks
    const int tm = wid / nchunks;                 // row tile (16 centers)
    const int ch = wid % nchunks;                 // column chunk

    const bool hi = (lane & 16) != 0;
    const int  l  = lane & 15;

    // A-matrix operand (hoisted): center row g, K pair via selects
    const int g = g0 + tm * 16 + l;
    const float* c = centers + (size_t)(b * NGRP + g) * 3;
    const float cx = c[0], cy = c[1], cz = c[2];
    v2f A;
    A.x = hi ? cz : cx;
    A.y = hi ? 0.0f : cy;

    // c2 for the 8 rows this lane's D registers cover (hoisted)
    const int rowloc = tm * 16 + (hi ? 8 : 0);
    float cc[8];
#pragma unroll
    for (int r = 0; r < 8; ++r)
        cc[r] = c2[(size_t)b * NGRP + g0 + rowloc + r];

    const float* xb = xyz + (size_t)b * NPTS * 3;

    for (int ti = 0; ti < CTILES; ++ti) {
        const int n = (ch * CTILES + ti) * 16 + l;
        const float* p = xb + (size_t)n * 3;
        const float px = p[0], py = p[1], pz = p[2];   // one b96 load, all lanes

        v2f Bm;
        Bm.x = hi ? pz : px;
        Bm.y = hi ? 0.0f : py;

        v8f C = {};
        v8f Dacc = __builtin_amdgcn_wmma_f32_16x16x4_f32(
            /*neg_a=*/false, A, /*neg_b=*/false, Bm,
            /*c_mod=*/(short)0, C, /*reuse_a=*/false, /*reuse_b=*/false);

        // C/D layout: VGPR r holds row M = r (lanes 0-15) / r+8 (lanes 16-31),
        // column N = lane & 15 in both halves.
        const float xx = px * px + py * py + pz * pz;  // |x|^2 inline
#pragma unroll
        for (int r = 0; r < 8; ++r) {
            const float val = cc[r] + xx - 2.0f * Dacc[r];
            slab[(size_t)(rowloc + r) * NPTS + n] = val;
        }
    }
}

// ---------------------------------------------------------------------------
// Kernel 4: top-32 selection + neighborhood gather. One block per group.
// Dist row staged to LDS via float4 (b128) loads; 32 rounds of lexicographic
// argmin (tie -> lower index), winner marked +inf. Reproduces
// lax.top_k(-dist2, 32) ordering exactly.
// ---------------------------------------------------------------------------
__global__ void select_kernel(const float* __restrict__ xyz,
                              const float* __restrict__ centers,
                              const float* __restrict__ slab,
                              float* __restrict__ out,
                              int b, int g0) {
    const int t  = threadIdx.x;
    const int gl = blockIdx.x;
    const int g  = g0 + gl;

    __shared__ float ds[NPTS];   // 32 KB
    __shared__ float rd[256];
    __shared__ int   ri[256];
    __shared__ int   sel[GSIZE];

    const float4* srow4 = (const float4*)(slab + (size_t)gl * NPTS);
    for (int i = t; i < NPTS / 4; i += 256) {
        const float4 v = srow4[i];
        ds[4 * i + 0] = v.x;
        ds[4 * i + 1] = v.y;
        ds[4 * i + 2] = v.z;
        ds[4 * i + 3] = v.w;
    }
    __syncthreads();

    for (int r = 0; r < GSIZE; ++r) {
        float bd = __builtin_inff();
        int   bi = 0x7fffffff;
        for (int i = t; i < NPTS; i += 256) {
            const float d = ds[i];
            if (d < bd || (d == bd && i < bi)) { bd = d; bi = i; }
        }
        rd[t] = bd;
        ri[t] = bi;
        __syncthreads();
        for (int off = 128; off > 0; off >>= 1) {
            if (t < off) {
                const float od = rd[t + off];
                const int   oi = ri[t + off];
                if (od < rd[t] || (od == rd[t] && oi < ri[t])) { rd[t] = od; ri[t] = oi; }
            }
            __syncthreads();
        }
        if (t == 0) {
            sel[r] = ri[0];
            ds[ri[0]] = __builtin_inff();
        }
        __syncthreads();
    }

    if (t < GSIZE) {
        const int idx = sel[t];
        const float* p = xyz + (size_t)(b * NPTS + idx) * 3;
        const float* c = centers + (size_t)(b * NGRP + g) * 3;
        float* o = out + ((size_t)(b * NGRP + g) * GSIZE + t) * 3;
        o[0] = p[0] - c[0];
        o[1] = p[1] - c[1];
        o[2] = p[2] - c[2];
    }
}

// ---------------------------------------------------------------------------
extern "C" void kernel_launch(void* const* d_in, const int* in_sizes, int n_in,
                              void* d_out, int out_size, void* d_ws, size_t ws_size,
                              hipStream_t stream) {
    (void)in_sizes; (void)n_in; (void)out_size;
    const float* xyz = (const float*)d_in[0];
    float* out = (float*)d_out;
    float* centers = out + (size_t)BATCH * NGRP * GSIZE * 3;

    float* c2   = (float*)d_ws;                       // B*G floats
    float* slab = c2 + (size_t)BATCH * NGRP;          // GC*N floats

    const size_t fixedFloats = (size_t)BATCH * NGRP;
    const size_t totalFloats = ws_size / sizeof(float);
    const size_t availFloats = totalFloats > fixedFloats ? totalFloats - fixedFloats : 0;

    int GC = NGRP;                                    // group-chunk: 512 -> 16
    while (GC > 16 && (size_t)GC * NPTS > availFloats) GC >>= 1;

    fps_kernel<<<BATCH, 1024, 0, stream>>>(xyz, out);
    prep_kernel<<<BATCH, 256, 0, stream>>>(centers, c2);

    for (int b = 0; b < BATCH; ++b) {
        for (int g0 = 0; g0 < NGRP; g0 += GC) {
            // one wave per (row tile, column chunk)
            const int waves  = (GC / 16) * (NPTS / (16 * CTILES));
            const int blocks = (waves + 7) / 8;
            dist_kernel<<<blocks, 256, 0, stream>>>(xyz, centers, c2, slab, b, g0);
            select_kernel<<<GC, 256, 0, stream>>>(xyz, centers, slab, out, b, g0);
        }
    }
}